// SparseICLModel_91311004713502
// MI455X (gfx1250) — compile-verified
//
#include <hip/hip_runtime.h>
#include <hip/hip_bf16.h>

// ---------------------------------------------------------------------------
// SparseICL transformer on gfx1250 (MI455X): all matmuls via V_WMMA_F32_16X16X4_F32
// (f32 tensor-core path => exact f32 numerics vs the reference).
// Seq = 4096, d = 128 (Q/K padded 127->128 with zeros), B = 4, L = 2.
// ---------------------------------------------------------------------------

typedef __attribute__((ext_vector_type(2))) float v2f;
typedef __attribute__((ext_vector_type(8))) float v8f;

#define SEQ    4096
#define DMODEL 128
#define NBATCH 4
#define NTOK   (NBATCH * SEQ)        // 16384 tokens total (batch-flattened)
#define MAX_SLOTS 72                 // max candidate key tiles per query tile (<=71)

// ---------------------------------------------------------------------------
// Build tokens: concat([xs, ys[...,None]], -1) then append [query_x, 0].
// ---------------------------------------------------------------------------
__global__ __launch_bounds__(256) void build_tokens(
    const float* __restrict__ xs, const float* __restrict__ ys,
    const float* __restrict__ qx, float* __restrict__ tok) {
  int idx = blockIdx.x * 256 + threadIdx.x;          // over NTOK*128
  if (idx >= NTOK * DMODEL) return;
  int c  = idx & 127;
  int ti = (idx >> 7) & (SEQ - 1);
  int b  = idx >> 19;
  float v;
  if (ti < SEQ - 1) {
    v = (c < 127) ? xs[((size_t)b * 4095 + ti) * 127 + c] : ys[(size_t)b * 4095 + ti];
  } else {
    v = (c < 127) ? qx[(size_t)b * 127 + c] : 0.0f;
  }
  tok[idx] = v;
}

// ---------------------------------------------------------------------------
// LayerNorm over last dim (128). One wave per token; optional residual addend.
// blockDim = 256 (8 waves = 8 tokens per block).
// ---------------------------------------------------------------------------
__global__ __launch_bounds__(256) void ln_kernel(
    const float* __restrict__ X, const float* __restrict__ add,
    const float* __restrict__ g, const float* __restrict__ beta,
    float* __restrict__ Y) {
  const int wave = threadIdx.x >> 5;
  const int lane = threadIdx.x & 31;
  const size_t tok = (size_t)blockIdx.x * 8 + wave;
  const size_t off = tok * DMODEL;

  float4 v = ((const float4*)(X + off))[lane];
  if (add != nullptr) {
    float4 a = ((const float4*)(add + off))[lane];
    v.x += a.x; v.y += a.y; v.z += a.z; v.w += a.w;
  }
  float sm = v.x + v.y + v.z + v.w;
  float sq = v.x * v.x + v.y * v.y + v.z * v.z + v.w * v.w;
  #pragma unroll
  for (int o = 16; o > 0; o >>= 1) {
    sm += __shfl_xor(sm, o, 32);
    sq += __shfl_xor(sq, o, 32);
  }
  const float mean = sm * (1.0f / 128.0f);
  const float var  = sq * (1.0f / 128.0f) - mean * mean;
  const float rs   = rsqrtf(var + 1e-5f);
  float4 gg = ((const float4*)g)[lane];
  float4 bb = ((const float4*)beta)[lane];
  float4 o;
  o.x = (v.x - mean) * rs * gg.x + bb.x;
  o.y = (v.y - mean) * rs * gg.y + bb.y;
  o.z = (v.z - mean) * rs * gg.z + bb.z;
  o.w = (v.w - mean) * rs * gg.w + bb.w;
  ((float4*)(Y + off))[lane] = o;
}

// ---------------------------------------------------------------------------
// Y = alpha * X @ W^T.  X: (NTOK x 128), W: (wrows x 128) row-major, Y: (NTOK x 128).
// Output columns n >= wrows are written as zero (pads Q/K hidden 127 -> 128).
// Branch-free inner loop: W row index clamped, invalid column zeroed at store.
// One wave per 16-row strip; per n-tile a fully unrolled 32-step WMMA chain
// whose B loads are all base+immediate (clusterable ahead of the WMMAs).
// ---------------------------------------------------------------------------
__global__ __launch_bounds__(32) void gemm_xwT(
    const float* __restrict__ X, const float* __restrict__ W,
    float* __restrict__ Y, int wrows, float alpha) {
  const int m0   = blockIdx.x * 16;
  const int lane = threadIdx.x & 31;
  const int mlo  = lane & 15;
  const int hi   = lane >> 4;
  const int koff = hi * 2;

  // ---- preload the 16x128 A strip in WMMA A-layout (32 x b64) -------------
  v2f qa[32];
  const float* xrow = X + (size_t)(m0 + mlo) * DMODEL + koff;
  #pragma unroll
  for (int ks = 0; ks < 32; ++ks)
    qa[ks] = *(const v2f*)(xrow + ks * 4);

  #pragma unroll
  for (int nt = 0; nt < 8; ++nt) {
    const int n  = nt * 16 + mlo;
    const int nc = (n < wrows) ? n : (wrows - 1);       // clamp: finite garbage
    const float* wr = W + (size_t)nc * DMODEL + koff;
    v8f acc = {};
    #pragma unroll
    for (int ks = 0; ks < 32; ++ks) {
      v2f b = *(const v2f*)(wr + ks * 4);
      acc = __builtin_amdgcn_wmma_f32_16x16x4_f32(
          false, qa[ks], false, b, (short)0, acc, false, false);
    }
    const float sel = (n < wrows) ? alpha : 0.0f;       // zero padded column
    #pragma unroll
    for (int r = 0; r < 8; ++r)
      Y[(size_t)(m0 + r + 8 * hi) * DMODEL + n] = acc[r] * sel;
  }
}

// ---------------------------------------------------------------------------
// Sparse-mask attention, one wave per (query tile Ti, batch b).
// Candidate key tiles: window {Ti-8..Ti} U stride diagonals {t == Ti (mod 4)} U {0}.
// Exact per-element mask applied, so superset enumeration is correct.
// S tiles staged in dynamic LDS (<= 71 KB), scalar row softmax, P*V via WMMA.
// ---------------------------------------------------------------------------
__device__ __forceinline__ bool tile_included(int t, int Ti) {
  return (t >= Ti - 8) || (((Ti - t) & 3) == 0) || (t == 0);
}

__global__ __launch_bounds__(32) void attn_sparse(
    const float* __restrict__ Q, const float* __restrict__ K,
    const float* __restrict__ V, float* __restrict__ O) {
  extern __shared__ float S[];                 // MAX_SLOTS * 256 floats
  const int Ti   = blockIdx.x;                 // 0..255 query tile
  const int b    = blockIdx.y;                 // 0..3
  const int base = b * SEQ;
  const int i0   = Ti * 16;
  const int lane = threadIdx.x & 31;
  const int mlo  = lane & 15;
  const int hi   = lane >> 4;
  const int koff = hi * 2;
  const float scale = 0.08873565094161139f;    // 1/sqrt(127)

  // ---- load the 16x128 Q tile into A-layout registers (32 x b64) ----------
  v2f qa[32];
  const float* qrow = Q + (size_t)(base + i0 + mlo) * DMODEL + koff;
  #pragma unroll
  for (int ks = 0; ks < 32; ++ks)
    qa[ks] = *(const v2f*)(qrow + ks * 4);

  // ---- phase 1: S = scale * Q K^T for each candidate tile, masked, to LDS -
  int s = 0;
  for (int t = 0; t <= Ti; ++t) {
    if (!tile_included(t, Ti)) continue;       // uniform branch (t, Ti uniform)
    v8f acc = {};
    const float* krow = K + (size_t)(base + t * 16 + mlo) * DMODEL + koff;
    #pragma unroll
    for (int ks = 0; ks < 32; ++ks) {
      v2f bb = *(const v2f*)(krow + ks * 4);
      acc = __builtin_amdgcn_wmma_f32_16x16x4_f32(
          false, qa[ks], false, bb, (short)0, acc, false, false);
    }
    #pragma unroll
    for (int r = 0; r < 8; ++r) {
      const int i = i0 + r + 8 * hi;
      const int j = t * 16 + mlo;
      const int d = i - j;
      const bool allowed = (d >= 0) && ((d < 128) || ((d & 63) == 0) || (j < 16));
      S[s * 256 + (r + 8 * hi) * 16 + mlo] = allowed ? acc[r] * scale : -3.0e38f;
    }
    ++s;
  }
  const int ntiles = s;
  __syncthreads();

  // ---- phase 2: row softmax over LDS (lane r owns query row r) ------------
  if (lane < 16) {
    float mx = -3.0e38f;
    for (int ss = 0; ss < ntiles; ++ss)
      for (int c = 0; c < 16; ++c)
        mx = fmaxf(mx, S[ss * 256 + lane * 16 + c]);
    float sum = 0.0f;
    for (int ss = 0; ss < ntiles; ++ss)
      for (int c = 0; c < 16; ++c) {
        const int idx = ss * 256 + lane * 16 + c;
        const float e = __expf(S[idx] - mx);
        S[idx] = e;
        sum += e;
      }
    const float inv = 1.0f / sum;
    for (int ss = 0; ss < ntiles; ++ss)
      for (int c = 0; c < 16; ++c)
        S[ss * 256 + lane * 16 + c] *= inv;
  }
  __syncthreads();

  // ---- phase 3: O = P V via WMMA (A from LDS, B strided from V) -----------
  v8f oacc[8] = {};
  s = 0;
  for (int t = 0; t <= Ti; ++t) {
    if (!tile_included(t, Ti)) continue;
    #pragma unroll
    for (int ks = 0; ks < 4; ++ks) {
      const int kk = ks * 4 + koff;            // key index within tile, 0..15
      v2f a = *(const v2f*)&S[s * 256 + mlo * 16 + kk];   // 8B-aligned ds_load_b64
      const float* v0 = V + (size_t)(base + t * 16 + kk) * DMODEL + mlo;
      const float* v1 = v0 + DMODEL;
      v2f bb[8];
      #pragma unroll
      for (int nt = 0; nt < 8; ++nt) {         // 16 loads clustered first
        bb[nt].x = v0[nt * 16];
        bb[nt].y = v1[nt * 16];
      }
      #pragma unroll
      for (int nt = 0; nt < 8; ++nt)
        oacc[nt] = __builtin_amdgcn_wmma_f32_16x16x4_f32(
            false, a, false, bb[nt], (short)0, oacc[nt], false, false);
    }
    ++s;
  }
  #pragma unroll
  for (int nt = 0; nt < 8; ++nt)
    #pragma unroll
    for (int r = 0; r < 8; ++r)
      O[(size_t)(base + i0 + r + 8 * hi) * DMODEL + nt * 16 + mlo] = oacc[nt][r];
}

// ---------------------------------------------------------------------------
// y[b] = tokens[b, -1, :] . Wp
// ---------------------------------------------------------------------------
__global__ __launch_bounds__(32) void predict_kernel(
    const float* __restrict__ tok, const float* __restrict__ wp,
    float* __restrict__ out) {
  const int b = blockIdx.x;
  const int lane = threadIdx.x;
  const float* t = tok + ((size_t)b * SEQ + (SEQ - 1)) * DMODEL;
  float4 v = ((const float4*)t)[lane];
  float4 w = ((const float4*)wp)[lane];
  float sacc = v.x * w.x + v.y * w.y + v.z * w.z + v.w * w.w;
  #pragma unroll
  for (int o = 16; o > 0; o >>= 1) sacc += __shfl_xor(sacc, o, 32);
  if (lane == 0) out[b] = sacc;
}

// ---------------------------------------------------------------------------
extern "C" void kernel_launch(void* const* d_in, const int* in_sizes, int n_in,
                              void* d_out, int out_size, void* d_ws, size_t ws_size,
                              hipStream_t stream) {
  const float* xs = (const float*)d_in[0];
  const float* ys = (const float*)d_in[1];
  const float* qx = (const float*)d_in[2];
  const float* Wq = (const float*)d_in[3];   // (L,127,128)
  const float* Wk = (const float*)d_in[4];   // (L,127,128)
  const float* Wv = (const float*)d_in[5];   // (L,128,128)
  const float* Wo = (const float*)d_in[6];   // (L,128,128)
  const float* g1 = (const float*)d_in[7];
  const float* b1 = (const float*)d_in[8];
  const float* g2 = (const float*)d_in[9];
  const float* b2 = (const float*)d_in[10];
  const float* Wp = (const float*)d_in[11];
  float* out = (float*)d_out;

  const size_t SZ = (size_t)NTOK * DMODEL;   // 2,097,152 floats per buffer
  float* ws  = (float*)d_ws;
  float* tok = ws;
  float* tn  = ws + 1 * SZ;                  // LN output; reused for O-proj output
  float* Qb  = ws + 2 * SZ;
  float* Kb  = ws + 3 * SZ;
  float* Vb  = ws + 4 * SZ;
  float* Ab  = ws + 5 * SZ;

  build_tokens<<<(NTOK * DMODEL + 255) / 256, 256, 0, stream>>>(xs, ys, qx, tok);

  for (int l = 0; l < 2; ++l) {
    ln_kernel<<<NTOK / 8, 256, 0, stream>>>(tok, nullptr, g1 + l * 128, b1 + l * 128, tn);
    gemm_xwT<<<NTOK / 16, 32, 0, stream>>>(tn, Wq + (size_t)l * 127 * 128, Qb, 127, 1.0f);
    gemm_xwT<<<NTOK / 16, 32, 0, stream>>>(tn, Wk + (size_t)l * 127 * 128, Kb, 127, 1.0f);
    gemm_xwT<<<NTOK / 16, 32, 0, stream>>>(tn, Wv + (size_t)l * 128 * 128, Vb, 128, 1.0f);
    dim3 agrid(SEQ / 16, NBATCH);
    attn_sparse<<<agrid, 32, MAX_SLOTS * 256 * sizeof(float), stream>>>(Qb, Kb, Vb, Ab);
    gemm_xwT<<<NTOK / 16, 32, 0, stream>>>(Ab, Wo + (size_t)l * 128 * 128, tn, 128, 0.1f);
    ln_kernel<<<NTOK / 8, 256, 0, stream>>>(tok, tn, g2 + l * 128, b2 + l * 128, tok);
  }

  predict_kernel<<<NBATCH, 32, 0, stream>>>(tok, Wp, out);
}